// Attention_17901423690229
// MI455X (gfx1250) — compile-verified
//
#include <hip/hip_runtime.h>
#include <cstdint>

// ---------------------------------------------------------------------------
// Segment attention for MI455X (gfx1250, wave32).
//  Per segment e (16 members):
//    A  = X[idx[e,0:16], :]                (16x128, f32, gathered)
//    Hm = leaky_relu(A @ W1 + b1)          (16x64)  -> V_WMMA_F32_16X16X4_F32
//    w  = Hm @ W2 (+b2, cancels in softmax) (16)
//    beta = softmax(w)                     (16)
//    Z  = tanh(leaky_relu(beta^T @ A))     (128)
//  Outputs flat-concatenated: [E*128 Z | T beta].
//
// Precision: fp32 WMMA (16x16x4) — workload is L2-resident (X = 51 MB < 192MB
// L2) and only ~13 GFLOP, so we keep full reference precision on the matrix
// pipe instead of chasing bf16 throughput we don't need.
// ---------------------------------------------------------------------------

typedef float v2f __attribute__((ext_vector_type(2)));
typedef float v8f __attribute__((ext_vector_type(8)));

#define C_DIM      128
#define H_DIM      64
#define E_SEG      50000
#define NEG_SLOPE  0.01f
#define WAVES_WG   8          // 256 threads = 8 wave32
#define SEGS_WAVE  2          // two segments per wave: B-fragments reused
#define W1_PITCH   132        // 128 + 4 pad: stride%64 == 4 -> 32 lanes x 2
                              // dwords cover all 64 LDS banks conflict-free

__global__ __launch_bounds__(256) void attn_seg_kernel(
    const float* __restrict__ X,          // (N_NODES, 128)
    const int*   __restrict__ member_idx, // (T)
    const float* __restrict__ W1,         // (128, 64) row-major
    const float* __restrict__ b1,         // (64)
    const float* __restrict__ W2,         // (64)
    float* __restrict__ outZ,             // (E, 128)
    float* __restrict__ outBeta)          // (T)
{
  __shared__ float lds_w1t[H_DIM * W1_PITCH];       // W1 transposed: [n][c]
  __shared__ float lds_b1[H_DIM];
  __shared__ float lds_w2[H_DIM];
  __shared__ float lds_beta[WAVES_WG * 16];
  __shared__ float lds_z[WAVES_WG * 128];

  const int tid = threadIdx.x;

  // Stage W1 transposed into LDS so a B fragment is one ds_load_b64.
  for (int i = tid; i < C_DIM * H_DIM; i += 256) {
    const int c = i >> 6;                  // 0..127 (K dim)
    const int n = i & 63;                  // 0..63  (N dim)
    lds_w1t[n * W1_PITCH + c] = W1[i];
  }
  if (tid < H_DIM) { lds_b1[tid] = b1[tid]; lds_w2[tid] = W2[tid]; }
  __syncthreads();

  const int wave = tid >> 5;
  const int lane = tid & 31;
  const int r    = lane & 15;             // A row / member index this lane owns
  const int kk   = (lane >> 4) << 1;      // K sub-offset: 0 (lanes 0-15) / 2

  // 3125 blocks * 8 waves * 2 segs == 50000 exactly -> no bounds checks,
  // all __syncthreads() are block-uniform.
  const int seg0 = (blockIdx.x * WAVES_WG + wave) * SEGS_WAVE;

  // ---- Gather A fragments for both segments (ISA 16x4 f32 A layout):
  //      lane (half*16 + r) holds row r, K = 4k+kk, 4k+kk+1  -> one float2.
  v2f a0[32], a1[32];
  {
    const int64_t n0 = (int64_t)member_idx[seg0 * 16 + r];
    const int64_t n1 = (int64_t)member_idx[(seg0 + 1) * 16 + r];
    const float* row0 = X + n0 * C_DIM;
    const float* row1 = X + n1 * C_DIM;
#pragma unroll
    for (int k = 0; k < 32; ++k) {
      a0[k] = *(const v2f*)(row0 + 4 * k + kk);
      a1[k] = *(const v2f*)(row1 + 4 * k + kk);
    }
  }

  // ---- H = A @ W1 : 4 output tiles (N=64) x 32 K-steps, fp32 WMMA.
  v8f acc0[4], acc1[4];
#pragma unroll
  for (int n = 0; n < 4; ++n) {
    acc0[n] = (v8f){0.f, 0.f, 0.f, 0.f, 0.f, 0.f, 0.f, 0.f};
    acc1[n] = (v8f){0.f, 0.f, 0.f, 0.f, 0.f, 0.f, 0.f, 0.f};
  }
#pragma unroll
  for (int k = 0; k < 32; ++k) {
    const int kbase = 4 * k + kk;          // B rows this lane supplies
#pragma unroll
    for (int n = 0; n < 4; ++n) {
      const int col = 16 * n + r;          // B column this lane supplies
      const v2f bf = *(const v2f*)&lds_w1t[col * W1_PITCH + kbase];
      acc0[n] = __builtin_amdgcn_wmma_f32_16x16x4_f32(
          false, a0[k], false, bf, (short)0, acc0[n], false, false);
      acc1[n] = __builtin_amdgcn_wmma_f32_16x16x4_f32(
          false, a1[k], false, bf, (short)0, acc1[n], false, false);
    }
  }

  // ---- Per-segment epilogue. C/D layout: VGPR j holds row m=j+8*(lane>=16),
  //      column N = 16n + r.
  auto epilogue = [&](v8f* acc, v2f* a, int seg) {
    float sj[8];
#pragma unroll
    for (int j = 0; j < 8; ++j) sj[j] = 0.f;
#pragma unroll
    for (int n = 0; n < 4; ++n) {
      const float bias = lds_b1[16 * n + r];
      const float w2v  = lds_w2[16 * n + r];
#pragma unroll
      for (int j = 0; j < 8; ++j) {
        float h = acc[n][j] + bias;
        h = (h > 0.f) ? h : h * NEG_SLOPE;          // leaky_relu
        sj[j] += h * w2v;                            // partial h @ W2
      }
    }
    // Reduce h@W2 over the 16 columns held by this half's lanes.
#pragma unroll
    for (int off = 1; off < 16; off <<= 1)
#pragma unroll
      for (int j = 0; j < 8; ++j) sj[j] += __shfl_xor(sj[j], off, 32);
    // sj[j] == w[m], m = j + 8*half.  (b2 shifts all w -> cancels in softmax.)

    // Segment softmax across 16 members (8 local + cross-half shuffle).
    float lmax = sj[0];
#pragma unroll
    for (int j = 1; j < 8; ++j) lmax = fmaxf(lmax, sj[j]);
    lmax = fmaxf(lmax, __shfl_xor(lmax, 16, 32));
    float ex[8], lsum = 0.f;
#pragma unroll
    for (int j = 0; j < 8; ++j) { ex[j] = __expf(sj[j] - lmax); lsum += ex[j]; }
    lsum += __shfl_xor(lsum, 16, 32);
    const float inv = 1.0f / lsum;
#pragma unroll
    for (int j = 0; j < 8; ++j) ex[j] *= inv;        // beta[m]

    float* bslot = &lds_beta[wave * 16];
    if (lane == 0 || lane == 16) {
      const int mbase = (lane >> 4) * 8;
#pragma unroll
      for (int j = 0; j < 8; ++j) {
        bslot[mbase + j] = ex[j];
        outBeta[seg * 16 + mbase + j] = ex[j];
      }
    }
    __syncthreads();
    const float br = bslot[r];                       // beta of this lane's row

    // Z[cols 4k+kk, 4k+kk+1] = sum over 16 rows of beta*A — reuse A regs.
    float* zslot = &lds_z[wave * 128];
#pragma unroll
    for (int k = 0; k < 32; ++k) {
      float zx = a[k].x * br;
      float zy = a[k].y * br;
#pragma unroll
      for (int off = 1; off < 16; off <<= 1) {
        zx += __shfl_xor(zx, off, 32);
        zy += __shfl_xor(zy, off, 32);
      }
      if (lane == 0 || lane == 16) {
        zslot[4 * k + kk]     = zx;
        zslot[4 * k + kk + 1] = zy;
      }
    }
    __syncthreads();

    // Coalesced finalize: each lane owns 4 contiguous channels.
    float4 z4 = *(const float4*)&zslot[lane * 4];
    float zx = z4.x > 0.f ? z4.x : z4.x * NEG_SLOPE;
    float zy = z4.y > 0.f ? z4.y : z4.y * NEG_SLOPE;
    float zz = z4.z > 0.f ? z4.z : z4.z * NEG_SLOPE;
    float zw = z4.w > 0.f ? z4.w : z4.w * NEG_SLOPE;
    float4 o4 = make_float4(tanhf(zx), tanhf(zy), tanhf(zz), tanhf(zw));
    *(float4*)(outZ + (size_t)seg * C_DIM + lane * 4) = o4;
    __syncthreads();   // slots reused next segment; keep block uniform
  };

  epilogue(acc0, a0, seg0);
  epilogue(acc1, a1, seg0 + 1);
}

extern "C" void kernel_launch(void* const* d_in, const int* in_sizes, int n_in,
                              void* d_out, int out_size, void* d_ws, size_t ws_size,
                              hipStream_t stream) {
  const float* X          = (const float*)d_in[0];
  const int*   member_idx = (const int*)d_in[1];
  // d_in[2] segment_ids: deterministic (t / 16), unused.
  const float* W1         = (const float*)d_in[3];
  const float* b1         = (const float*)d_in[4];
  const float* W2         = (const float*)d_in[5];
  // d_in[6] b2: constant shift of pre-softmax logits -> cancels; unused.

  float* outZ    = (float*)d_out;                          // E*128
  float* outBeta = (float*)d_out + (size_t)E_SEG * C_DIM;  // T

  const int chunks = E_SEG / SEGS_WAVE;          // 25000 wave-chunks
  const int blocks = chunks / WAVES_WG;          // 3125 (exact)
  attn_seg_kernel<<<blocks, 256, 0, stream>>>(X, member_idx, W1, b1, W2,
                                              outZ, outBeta);
}